// Decoder_89928025244439
// MI455X (gfx1250) — compile-verified
//
#include <hip/hip_runtime.h>
#include <cstddef>

// ---------------------------------------------------------------------------
// 2-layer bidirectional LSTM + per-timestep Linear for MI455X (gfx1250).
//
// Persistent per-layer kernels, weights stationary in LDS across all 1024
// timesteps, cell state c resident in LDS, hidden state h circulated through
// L2 with a software grid barrier each step. All matrix work uses
// V_WMMA_F32_16X16X4_F32 (native fp32 WMMA) to match the fp32 reference.
// Dual accumulators per wave break the WMMA RAW dependency chain.
// ---------------------------------------------------------------------------

typedef float v2f __attribute__((ext_vector_type(2)));
typedef float v8f __attribute__((ext_vector_type(8)));

// D = A(16x4) * B(4x16) + C, fp32 throughout.
// A per lane: lanes 0-15 -> M=lane, K=k0+0/+1; lanes 16-31 -> M=lane-16, K=k0+2/+3
// B per lane: lanes 0-15 -> N=lane,  K=k0+0/+1; lanes 16-31 -> N=lane-16, K=k0+2/+3
// C/D: VGPR v: lanes 0-15 -> (M=v,   N=lane); lanes 16-31 -> (M=v+8, N=lane-16)
__device__ __forceinline__ v8f wmma_f32(v2f a, v2f b, v8f c) {
  return __builtin_amdgcn_wmma_f32_16x16x4_f32(false, a, false, b, (short)0, c,
                                               false, false);
}

__device__ __forceinline__ float sigm_f(float x) {
  return 1.0f / (1.0f + __expf(-x));   // exp(-x)->inf underflows to 0 safely
}
__device__ __forceinline__ float tanh_f(float x) {
  // tanh(x) = sign(x) * (1 - e^{-2|x|}) / (1 + e^{-2|x|}); e in (0,1] -> safe
  float a = __expf(-2.0f * fabsf(x));
  float r = (1.0f - a) / (1.0f + a);
  return copysignf(r, x);
}

constexpr int B_ = 32;      // batch
constexpr int T_ = 1024;    // sequence length
constexpr int H_ = 512;     // hidden
constexpr int G_ = 4 * H_;  // 2048 gate columns
constexpr int O_ = 256;     // output dim

// ---------------------------------------------------------------------------
// init: zero the grid-barrier counters (workspace is poisoned with 0xAA and
// not re-poisoned between graph replays, so we must reset them every call).
// ---------------------------------------------------------------------------
__global__ void init_kernel(unsigned* bar) {
  if (threadIdx.x < 16) bar[threadIdx.x] = 0u;
}

// ---------------------------------------------------------------------------
// Persistent bidirectional LSTM layer.
//   grid = 64 WGs: blockIdx>>5 = direction (0 fwd, 1 bwd), blockIdx&31 = slice.
//   Each WG owns hidden units [j0, j0+16) of one direction:
//     - gate columns n = g*512 + j0 + col for g in {i,f,g,o}  (4 N-tiles)
//     - batch 32 -> 2 M-tiles; 8 waves -> one 16x16 C-tile per wave.
//   LDS: [64][WST] weight rows (B-operand, stationary all steps; WST padded so
//        row stride % 64 == 4 -> conflict-free ds_load_b64), 4x32x16 gate
//        exchange buffer, 32x16 resident cell state.
//   Per step: dual-accumulator WMMA over K (input K_IN + hidden 512), gate
//   exchange via LDS, elementwise LSTM update, publish h, grid barrier.
// ---------------------------------------------------------------------------
template <int K_IN, bool WIH_LDS>
__global__ void lstm_layer_kernel(
    const float* __restrict__ inp,    // [B, T, K_IN]
    const float* __restrict__ w_ih,   // [2, 2048, K_IN]
    const float* __restrict__ w_hh,   // [2, 2048, 512]
    const float* __restrict__ b_ih,   // [2, 2048]
    const float* __restrict__ b_hh,   // [2, 2048]
    const float* __restrict__ h0,     // [2, 32, 512] (this layer's slice)
    const float* __restrict__ c0,     // [2, 32, 512]
    float* __restrict__ h_buf,        // ws: [2 ping-pong][2 dir][32][512]
    unsigned* __restrict__ bar,       // ws: monotonic barrier counter
    float* __restrict__ y_out,        // [B, T, 1024] (fwd 0-511, bwd 512-1023)
    float* __restrict__ hh_out,       // [4, 32, 512]
    float* __restrict__ hc_out,       // [4, 32, 512]
    int layer) {
  const int tid = threadIdx.x;
  const int lane = tid & 31;
  const int wave = tid >> 5;
  const int dir = blockIdx.x >> 5;
  const int j0 = (blockIdx.x & 31) * 16;
  const int gate = wave >> 1;        // 0=i 1=f 2=g 3=o
  const int m0v = (wave & 1) * 16;   // M-tile base (batch)
  const int nWG = gridDim.x;

  constexpr int WROWLEN = WIH_LDS ? (K_IN + 512) : 512;  // valid floats/row
  constexpr int WST = WROWLEN + 4;   // stride%64==4 -> conflict-free banks
  constexpr int HOFF = WIH_LDS ? K_IN : 0;

  extern __shared__ float smem[];
  float* wlds = smem;                 // [64][WST]
  float* glds = smem + 64 * WST;      // [4][32][16] gate exchange
  float* clds = glds + 4 * 32 * 16;   // [32][16] resident cell state

  // ---- one-time: stationary weights (B-operand rows) into LDS -------------
  for (int idx = tid; idx < 64 * WROWLEN; idx += 256) {
    int r = idx / WROWLEN, k = idx - r * WROWLEN;
    int n = (r >> 4) * 512 + j0 + (r & 15);
    float v;
    if (WIH_LDS)
      v = (k < K_IN) ? w_ih[(size_t)(dir * G_ + n) * K_IN + k]
                     : w_hh[(size_t)(dir * G_ + n) * 512 + (k - K_IN)];
    else
      v = w_hh[(size_t)(dir * G_ + n) * 512 + k];
    wlds[r * WST + k] = v;
  }
  // ---- one-time: resident cell state --------------------------------------
  for (int idx = tid; idx < 512; idx += 256) {
    int m = idx >> 4, j = idx & 15;
    clds[idx] = c0[(dir * 32 + m) * 512 + j0 + j];
  }
  __syncthreads();

  // per-lane constants for WMMA fragments
  const int ncol = gate * 512 + j0 + (lane & 15);          // gate column (B/N)
  const float bias = b_ih[dir * G_ + ncol] + b_hh[dir * G_ + ncol];
  const int am = m0v + (lane & 15);                        // batch row (A/M)
  const int koff = (lane >> 4) << 1;                       // +0 or +2 in K
  const float* brow_lds = wlds + (gate * 16 + (lane & 15)) * WST;
  const float* brow_gbl = w_ih + (size_t)(dir * G_ + ncol) * K_IN;

  for (int s = 0; s < T_; ++s) {
    const int t = dir ? (T_ - 1 - s) : s;
    const float* hsrc = (s == 0)
        ? (h0 + dir * 32 * 512)
        : (h_buf + (size_t)(s & 1) * (2 * 32 * 512) + dir * 32 * 512);

    // Dual accumulators: even K-chunks -> acc0 (seeded with bias),
    // odd K-chunks -> acc1 (zero). Breaks the WMMA RAW chain in half.
    v8f acc0, acc1;
#pragma unroll
    for (int v = 0; v < 8; ++v) { acc0[v] = bias; acc1[v] = 0.0f; }

    // ---- input contribution: K = K_IN (multiple of 8) ---------------------
    const float* arow = inp + ((size_t)am * T_ + t) * K_IN;
#pragma unroll 2
    for (int k0 = 0; k0 < K_IN; k0 += 8) {
      v2f a0 = *(const v2f*)(arow + k0 + koff);
      v2f a1 = *(const v2f*)(arow + k0 + 4 + koff);
      v2f b0, b1;
      if (WIH_LDS) {
        b0 = *(const v2f*)(brow_lds + k0 + koff);
        b1 = *(const v2f*)(brow_lds + k0 + 4 + koff);
      } else {
        b0 = *(const v2f*)(brow_gbl + k0 + koff);
        b1 = *(const v2f*)(brow_gbl + k0 + 4 + koff);
      }
      acc0 = wmma_f32(a0, b0, acc0);
      acc1 = wmma_f32(a1, b1, acc1);
    }
    // ---- recurrent contribution: K = 512 ----------------------------------
    const float* hrow = hsrc + am * 512;
#pragma unroll 2
    for (int k0 = 0; k0 < 512; k0 += 8) {
      v2f a0 = *(const v2f*)(hrow + k0 + koff);
      v2f a1 = *(const v2f*)(hrow + k0 + 4 + koff);
      v2f b0 = *(const v2f*)(brow_lds + HOFF + k0 + koff);
      v2f b1 = *(const v2f*)(brow_lds + HOFF + k0 + 4 + koff);
      acc0 = wmma_f32(a0, b0, acc0);
      acc1 = wmma_f32(a1, b1, acc1);
    }

    // ---- exchange gate tiles through LDS ----------------------------------
    {
      int col = lane & 15;
      int mbase = m0v + ((lane >> 4) << 3);  // lanes 16-31 hold M rows +8
#pragma unroll
      for (int v = 0; v < 8; ++v)
        glds[(gate * 32 + mbase + v) * 16 + col] = acc0[v] + acc1[v];
    }
    __syncthreads();

    // ---- elementwise LSTM cell: 512 (m,j) pairs, 2 per thread -------------
    float* hdst =
        h_buf + (size_t)((s + 1) & 1) * (2 * 32 * 512) + dir * 32 * 512;
#pragma unroll
    for (int u = 0; u < 2; ++u) {
      int idx = tid + u * 256;
      int m = idx >> 4, j = idx & 15;
      float gi = glds[(0 * 32 + m) * 16 + j];
      float gf = glds[(1 * 32 + m) * 16 + j];
      float gg = glds[(2 * 32 + m) * 16 + j];
      float go = glds[(3 * 32 + m) * 16 + j];
      float c = sigm_f(gf) * clds[idx] + sigm_f(gi) * tanh_f(gg);
      float h = sigm_f(go) * tanh_f(c);
      clds[idx] = c;
      hdst[m * 512 + j0 + j] = h;
      y_out[((size_t)m * T_ + t) * 1024 + dir * 512 + j0 + j] = h;
      if (s == T_ - 1) {  // final carry of the scan (fwd: t=T-1, bwd: t=0)
        hh_out[((layer * 2 + dir) * 32 + m) * 512 + j0 + j] = h;
        hc_out[((layer * 2 + dir) * 32 + m) * 512 + j0 + j] = c;
      }
    }

    // ---- grid-wide barrier (h must be visible to all 64 WGs) --------------
    __threadfence();
    __syncthreads();
    if (tid == 0) {
      __hip_atomic_fetch_add(bar, 1u, __ATOMIC_RELEASE,
                             __HIP_MEMORY_SCOPE_AGENT);
      const unsigned target = (unsigned)(s + 1) * (unsigned)nWG;
      while (__hip_atomic_load(bar, __ATOMIC_ACQUIRE,
                               __HIP_MEMORY_SCOPE_AGENT) < target)
        __builtin_amdgcn_s_sleep(1);
    }
    __syncthreads();
    __threadfence();
  }
}

// ---------------------------------------------------------------------------
// outputs[m, n] = y1[m, :1024] @ w_out[n, :1024] + b_out[n]
// M = B*T = 32768, N = 256, K = 1024. 8 waves/WG -> 64x32 C-block per WG.
// ---------------------------------------------------------------------------
__global__ void out_linear_kernel(const float* __restrict__ y1,    // [M,1024]
                                  const float* __restrict__ w_out, // [256,1024]
                                  const float* __restrict__ b_out, // [256]
                                  float* __restrict__ out) {       // [M,256]
  const int tid = threadIdx.x;
  const int lane = tid & 31;
  const int wave = tid >> 5;
  const int m0 = blockIdx.x * 64 + (wave & 3) * 16;
  const int n0 = blockIdx.y * 32 + (wave >> 2) * 16;
  const int col = lane & 15;
  const int koff = (lane >> 4) << 1;
  const int am = m0 + col;
  const int nc = n0 + col;

  v8f acc0, acc1;
  const float bias = b_out[nc];
#pragma unroll
  for (int v = 0; v < 8; ++v) { acc0[v] = bias; acc1[v] = 0.0f; }

  const float* arow = y1 + (size_t)am * 1024;
  const float* brow = w_out + (size_t)nc * 1024;
#pragma unroll 4
  for (int k0 = 0; k0 < 1024; k0 += 8) {
    v2f a0 = *(const v2f*)(arow + k0 + koff);
    v2f b0 = *(const v2f*)(brow + k0 + koff);
    v2f a1 = *(const v2f*)(arow + k0 + 4 + koff);
    v2f b1 = *(const v2f*)(brow + k0 + 4 + koff);
    acc0 = wmma_f32(a0, b0, acc0);
    acc1 = wmma_f32(a1, b1, acc1);
  }

  const int mbase = m0 + ((lane >> 4) << 3);
#pragma unroll
  for (int v = 0; v < 8; ++v)
    out[(size_t)(mbase + v) * O_ + nc] = acc0[v] + acc1[v];
}

// ---------------------------------------------------------------------------
// Host launcher.
// ---------------------------------------------------------------------------
extern "C" void kernel_launch(void* const* d_in, const int* in_sizes, int n_in,
                              void* d_out, int out_size, void* d_ws,
                              size_t ws_size, hipStream_t stream) {
  (void)in_sizes; (void)n_in; (void)out_size; (void)ws_size;

  const float* x        = (const float*)d_in[0];   // [32,1024,256]
  const float* h0       = (const float*)d_in[1];   // [4,32,512]
  const float* c0       = (const float*)d_in[2];   // [4,32,512]
  const float* w_ih_l0  = (const float*)d_in[3];   // [2,2048,256]
  const float* w_hh_l0  = (const float*)d_in[4];   // [2,2048,512]
  const float* b_ih_l0  = (const float*)d_in[5];   // [2,2048]
  const float* b_hh_l0  = (const float*)d_in[6];   // [2,2048]
  const float* w_ih_l1  = (const float*)d_in[7];   // [2,2048,1024]
  const float* w_hh_l1  = (const float*)d_in[8];   // [2,2048,512]
  const float* b_ih_l1  = (const float*)d_in[9];   // [2,2048]
  const float* b_hh_l1  = (const float*)d_in[10];  // [2,2048]
  const float* w_out    = (const float*)d_in[11];  // [256,1024]
  const float* b_out    = (const float*)d_in[12];  // [256]

  // workspace: [0..63] barrier counters | h ping-pong [2][2][32][512] | y0
  float*    ws    = (float*)d_ws;
  unsigned* bar   = (unsigned*)d_ws;
  float*    h_buf = ws + 64;
  float*    y0    = h_buf + 2 * 2 * 32 * 512;      // [32,1024,1024] = 128 MB

  // d_out: outputs | y1 | hidden_h | hidden_c (flat, tuple return order)
  float* outputs = (float*)d_out;
  float* y1 = outputs + (size_t)B_ * T_ * O_;          // [32,1024,1024]
  float* hh = y1 + (size_t)B_ * T_ * 1024;             // [4,32,512]
  float* hc = hh + (size_t)4 * B_ * H_;                // [4,32,512]

  init_kernel<<<1, 64, 0, stream>>>(bar);

  // layer 0: W_ih||W_hh fully LDS-resident: 64*(768+4)*4 + (2048+512)*4 B
  const size_t lds0 = (size_t)(64 * (256 + 512 + 4) + 4 * 32 * 16 + 512) * 4;
  lstm_layer_kernel<256, true><<<64, 256, lds0, stream>>>(
      x, w_ih_l0, w_hh_l0, b_ih_l0, b_hh_l0,
      h0, c0, h_buf, bar + 0, y0, hh, hc, /*layer=*/0);

  // layer 1: only W_hh in LDS (W_ih_l1 streamed from L2-resident global)
  const size_t lds1 = (size_t)(64 * (512 + 4) + 4 * 32 * 16 + 512) * 4;
  lstm_layer_kernel<1024, false><<<64, 256, lds1, stream>>>(
      y0, w_ih_l1, w_hh_l1, b_ih_l1, b_hh_l1,
      h0 + 2 * 32 * 512, c0 + 2 * 32 * 512, h_buf, bar + 1, y1, hh, hc,
      /*layer=*/1);

  // final per-timestep Linear: [32768,1024] x [1024,256]
  out_linear_kernel<<<dim3(512, 8), 256, 0, stream>>>(y1, w_out, b_out,
                                                      outputs);
}